// Top2Gate_48369921688084
// MI455X (gfx1250) — compile-verified
//
#include <hip/hip_runtime.h>
#include <math.h>

#define T_TOK 8192
#define NE    64
#define DIM   2048
#define CAP   256
#define TEC   ((size_t)T_TOK * NE * CAP)   // 134,217,728 elements per output tensor

typedef __attribute__((ext_vector_type(2))) float v2f;
typedef __attribute__((ext_vector_type(4))) float v4f;
typedef __attribute__((ext_vector_type(8))) float v8f;

// ---------------------------------------------------------------------------
// Kernel 0: zero the giant output (1 + 2*TEC floats). 128-bit stores.
// grid 32768 x 256, 8 v4f per thread -> 268,435,456 floats + 1 tail scalar.
// ---------------------------------------------------------------------------
__global__ __launch_bounds__(256) void zero_out(float* __restrict__ out) {
  const size_t nthreads = (size_t)gridDim.x * 256;
  const size_t idx = (size_t)blockIdx.x * 256 + threadIdx.x;
  v4f z = {};
#pragma unroll
  for (int j = 0; j < 8; ++j) {
    size_t p = idx + (size_t)j * nthreads;
    *(v4f*)(out + p * 4) = z;
  }
  if (idx == 0) out[2 * TEC] = 0.0f;   // last element (index 268,435,456)
}

// ---------------------------------------------------------------------------
// Kernel 1: logits[T,E] = x[T,D] @ wg[E,D]^T via V_WMMA_F32_16X16X4_F32.
// 64 blocks x 8 waves; wave owns one 16-token M tile and all four 16-expert
// N tiles, so x is read exactly once (64 MB); wg (0.5 MB) hits L2/WGP$.
// ISA A layout (16x4 f32): lanes 0-15 hold K=0,1 ; lanes 16-31 hold K=2,3
// -> per-lane contiguous float2 at x[m, k + 2*(lane>>4)]. B mirrors it.
// ---------------------------------------------------------------------------
__global__ __launch_bounds__(256) void gemm_logits(const float* __restrict__ x,
                                                   const float* __restrict__ wg,
                                                   float* __restrict__ logits) {
  const int lane = threadIdx.x & 31;
  const int wave = threadIdx.x >> 5;
  const int m0   = (blockIdx.x * 8 + wave) * 16;
  const int row  = lane & 15;
  const int kh   = lane >> 4;

  v8f acc0 = {}, acc1 = {}, acc2 = {}, acc3 = {};

  const float* xr = x  + (size_t)(m0 + row) * DIM + 2 * kh;
  const float* w0 = wg + (size_t)( 0 + row) * DIM + 2 * kh;
  const float* w1 = wg + (size_t)(16 + row) * DIM + 2 * kh;
  const float* w2 = wg + (size_t)(32 + row) * DIM + 2 * kh;
  const float* w3 = wg + (size_t)(48 + row) * DIM + 2 * kh;

#pragma unroll 4
  for (int k = 0; k < DIM; k += 4) {
    v2f a  = *(const v2f*)(xr + k);
    v2f b0 = *(const v2f*)(w0 + k);
    v2f b1 = *(const v2f*)(w1 + k);
    v2f b2 = *(const v2f*)(w2 + k);
    v2f b3 = *(const v2f*)(w3 + k);
    acc0 = __builtin_amdgcn_wmma_f32_16x16x4_f32(false, a, false, b0, (short)0, acc0, false, false);
    acc1 = __builtin_amdgcn_wmma_f32_16x16x4_f32(false, a, false, b1, (short)0, acc1, false, false);
    acc2 = __builtin_amdgcn_wmma_f32_16x16x4_f32(false, a, false, b2, (short)0, acc2, false, false);
    acc3 = __builtin_amdgcn_wmma_f32_16x16x4_f32(false, a, false, b3, (short)0, acc3, false, false);
  }

  // C/D layout: VGPR r -> M = r + 8*(lane>>4), N = lane&15 (per 16-lane half)
#pragma unroll
  for (int r = 0; r < 8; ++r) {
    float* o = logits + (size_t)(m0 + r + 8 * kh) * NE + row;
    o[0]  = acc0[r];
    o[16] = acc1[r];
    o[32] = acc2[r];
    o[48] = acc3[r];
  }
}

// ---------------------------------------------------------------------------
// Kernel 2: per-token softmax over 64 experts + top1 + gumbel-masked top2.
// One wave per token, 2 experts per lane, shfl_xor butterfly reductions.
// ---------------------------------------------------------------------------
__global__ __launch_bounds__(256) void softmax_top2(const float* __restrict__ logits,
                                                    const float* __restrict__ gumbel,
                                                    float* __restrict__ gates,
                                                    int* __restrict__ idx1,
                                                    int* __restrict__ idx2,
                                                    float* __restrict__ g1,
                                                    float* __restrict__ g2) {
  const int lane = threadIdx.x & 31;
  const int t    = blockIdx.x * 8 + (threadIdx.x >> 5);

  const float* lrow = logits + (size_t)t * NE;
  const float* grow = gumbel + (size_t)t * NE;
  v2f L = *(const v2f*)(lrow + 2 * lane);
  v2f G = *(const v2f*)(grow + 2 * lane);

  // wave max
  float m = fmaxf(L.x, L.y);
  for (int off = 16; off; off >>= 1) m = fmaxf(m, __shfl_xor(m, off, 32));
  // softmax denom
  float e0 = __expf(L.x - m), e1 = __expf(L.y - m);
  float s = e0 + e1;
  for (int off = 16; off; off >>= 1) s += __shfl_xor(s, off, 32);
  float inv = 1.0f / s;

  v2f gv; gv.x = e0 * inv; gv.y = e1 * inv;
  *(v2f*)(gates + (size_t)t * NE + 2 * lane) = gv;

  // argmax of logits (== argmax of gates), first-max tie-break (lowest index)
  float v; int i;
  if (L.x >= L.y) { v = L.x; i = 2 * lane; } else { v = L.y; i = 2 * lane + 1; }
  for (int off = 16; off; off >>= 1) {
    float ov = __shfl_xor(v, off, 32);
    int   oi = __shfl_xor(i, off, 32);
    if (ov > v || (ov == v && oi < i)) { v = ov; i = oi; }
  }
  const int best1 = i;
  const float gate1 = __expf(v - m) * inv;

  // second expert: argmax over gumbel-noised logits with top-1 masked to -inf
  float n0 = (2 * lane     == best1) ? -INFINITY : L.x + G.x;
  float n1 = (2 * lane + 1 == best1) ? -INFINITY : L.y + G.y;
  float v2; int i2; float a2;   // a2 carries the original logit for gate value
  if (n0 >= n1) { v2 = n0; i2 = 2 * lane;     a2 = L.x; }
  else          { v2 = n1; i2 = 2 * lane + 1; a2 = L.y; }
  for (int off = 16; off; off >>= 1) {
    float ov = __shfl_xor(v2, off, 32);
    int   oi = __shfl_xor(i2, off, 32);
    float oa = __shfl_xor(a2, off, 32);
    if (ov > v2 || (ov == v2 && oi < i2)) { v2 = ov; i2 = oi; a2 = oa; }
  }
  const float gate2 = __expf(a2 - m) * inv;

  if (lane == 0) { idx1[t] = best1; idx2[t] = i2; g1[t] = gate1; g2[t] = gate2; }
}

// ---------------------------------------------------------------------------
// Kernel 3a: per-chunk expert histograms (64 chunks x 128 tokens).
// Integer LDS atomics -> deterministic counts.
// ---------------------------------------------------------------------------
__global__ __launch_bounds__(128) void chunk_hist(const int* __restrict__ idx1,
                                                  const int* __restrict__ idx2,
                                                  int* __restrict__ hist1,
                                                  int* __restrict__ hist2) {
  __shared__ int h1[NE], h2[NE];
  const int tid = threadIdx.x;
  if (tid < NE) { h1[tid] = 0; h2[tid] = 0; }
  __syncthreads();
  const int t = blockIdx.x * 128 + tid;
  atomicAdd(&h1[idx1[t]], 1);
  atomicAdd(&h2[idx2[t]], 1);
  __syncthreads();
  if (tid < NE) {
    hist1[blockIdx.x * NE + tid] = h1[tid];
    hist2[blockIdx.x * NE + tid] = h2[tid];
  }
}

// ---------------------------------------------------------------------------
// Kernel 3b: exclusive prefix over chunks per expert. loc2 offsets start at
// total mask1 count (locations2 = cumsum(mask2)-1 + sum(mask1)).
// ---------------------------------------------------------------------------
__global__ __launch_bounds__(64) void expert_offsets(const int* __restrict__ hist1,
                                                     const int* __restrict__ hist2,
                                                     int* __restrict__ off1,
                                                     int* __restrict__ off2,
                                                     int* __restrict__ cnt1) {
  const int e = threadIdx.x;
  int r = 0;
  for (int c = 0; c < 64; ++c) { off1[c * NE + e] = r; r += hist1[c * NE + e]; }
  cnt1[e] = r;
  int r2 = r;
  for (int c = 0; c < 64; ++c) { off2[c * NE + e] = r2; r2 += hist2[c * NE + e]; }
}

// ---------------------------------------------------------------------------
// Kernel 3c: ordered in-chunk ranks. Thread = expert, sequential over the
// 128-token chunk -> exact cumsum semantics, fully deterministic.
// ---------------------------------------------------------------------------
__global__ __launch_bounds__(64) void chunk_rank(const int* __restrict__ idx1,
                                                 const int* __restrict__ idx2,
                                                 const int* __restrict__ off1,
                                                 const int* __restrict__ off2,
                                                 int* __restrict__ loc1,
                                                 int* __restrict__ loc2) {
  __shared__ int s1[128], s2[128];
  const int tid  = threadIdx.x;
  const int base = blockIdx.x * 128;
  s1[tid]      = idx1[base + tid];
  s1[tid + 64] = idx1[base + tid + 64];
  s2[tid]      = idx2[base + tid];
  s2[tid + 64] = idx2[base + tid + 64];
  __syncthreads();
  const int e = tid;
  int b1 = off1[blockIdx.x * NE + e];
  int b2 = off2[blockIdx.x * NE + e];
  for (int i = 0; i < 128; ++i) {
    if (s1[i] == e) loc1[base + i] = b1++;
    if (s2[i] == e) loc2[base + i] = b2++;
  }
}

// ---------------------------------------------------------------------------
// Kernel 4: l_aux = E^2 * mean(me * ce) = 64 * sum_e me[e]*ce[e].
// Fixed-order per-thread partial sums + LDS tree -> deterministic.
// ---------------------------------------------------------------------------
__global__ __launch_bounds__(256) void me_laux(const float* __restrict__ gates,
                                               const int* __restrict__ cnt1,
                                               float* __restrict__ dout) {
  __shared__ float part[256];
  __shared__ float red[64];
  const int tid = threadIdx.x;
  const int e = tid & 63;
  const int q = tid >> 6;           // 4 token quarters of 2048
  float s = 0.0f;
  const int t0 = q * 2048;
  for (int i = 0; i < 2048; ++i) s += gates[(size_t)(t0 + i) * NE + e];
  part[tid] = s;
  __syncthreads();
  if (tid < 64) {
    float tot = part[e] + part[64 + e] + part[128 + e] + part[192 + e];
    float me = tot * (1.0f / T_TOK);
    float ce = (float)cnt1[e] * (1.0f / T_TOK);
    red[tid] = me * ce;
  }
  __syncthreads();
  for (int st = 32; st > 0; st >>= 1) {
    if (tid < st) red[tid] += red[tid + st];
    __syncthreads();
  }
  if (tid == 0) dout[0] = 64.0f * red[0];
}

// ---------------------------------------------------------------------------
// Kernel 5: capacity drop, renormalize, scatter into combine + dispatch.
// ---------------------------------------------------------------------------
__global__ __launch_bounds__(256) void scatter_out(const int* __restrict__ idx1,
                                                   const int* __restrict__ idx2,
                                                   const int* __restrict__ loc1,
                                                   const int* __restrict__ loc2,
                                                   const float* __restrict__ g1,
                                                   const float* __restrict__ g2,
                                                   float* __restrict__ dout) {
  const int t = blockIdx.x * 256 + threadIdx.x;
  const int e1 = idx1[t], e2 = idx2[t];
  const int L1 = loc1[t], L2 = loc2[t];
  const bool k1 = L1 < CAP, k2 = L2 < CAP;
  const float ga = k1 ? g1[t] : 0.0f;
  const float gb = k2 ? g2[t] : 0.0f;
  const float denom = fmaxf(ga + gb, 1.1920929e-07f);   // float32 eps
  const float c1 = ga / denom, c2 = gb / denom;
  float* comb = dout + 1;
  float* disp = dout + 1 + TEC;
  if (k1) {
    size_t o = (size_t)t * NE * CAP + (size_t)e1 * CAP + L1;
    comb[o] = c1; disp[o] = 1.0f;
  }
  if (k2) {
    size_t o = (size_t)t * NE * CAP + (size_t)e2 * CAP + L2;
    comb[o] = c2; disp[o] = 1.0f;
  }
}

// ---------------------------------------------------------------------------
extern "C" void kernel_launch(void* const* d_in, const int* in_sizes, int n_in,
                              void* d_out, int out_size, void* d_ws, size_t ws_size,
                              hipStream_t stream) {
  (void)in_sizes; (void)n_in; (void)out_size; (void)ws_size;
  const float* x      = (const float*)d_in[0];
  const float* wg     = (const float*)d_in[1];
  const float* gumbel = (const float*)d_in[2];
  float* out = (float*)d_out;

  char* w = (char*)d_ws;
  float* logits = (float*)w; w += (size_t)T_TOK * NE * 4;
  float* gates  = (float*)w; w += (size_t)T_TOK * NE * 4;
  int*   idx1   = (int*)w;   w += (size_t)T_TOK * 4;
  int*   idx2   = (int*)w;   w += (size_t)T_TOK * 4;
  float* g1v    = (float*)w; w += (size_t)T_TOK * 4;
  float* g2v    = (float*)w; w += (size_t)T_TOK * 4;
  int*   loc1   = (int*)w;   w += (size_t)T_TOK * 4;
  int*   loc2   = (int*)w;   w += (size_t)T_TOK * 4;
  int*   hist1  = (int*)w;   w += (size_t)64 * NE * 4;
  int*   hist2  = (int*)w;   w += (size_t)64 * NE * 4;
  int*   off1   = (int*)w;   w += (size_t)64 * NE * 4;
  int*   off2   = (int*)w;   w += (size_t)64 * NE * 4;
  int*   cnt1   = (int*)w;   w += (size_t)NE * 4;

  zero_out      <<<32768, 256, 0, stream>>>(out);
  gemm_logits   <<<64,    256, 0, stream>>>(x, wg, logits);
  softmax_top2  <<<1024,  256, 0, stream>>>(logits, gumbel, gates, idx1, idx2, g1v, g2v);
  chunk_hist    <<<64,    128, 0, stream>>>(idx1, idx2, hist1, hist2);
  expert_offsets<<<1,      64, 0, stream>>>(hist1, hist2, off1, off2, cnt1);
  chunk_rank    <<<64,     64, 0, stream>>>(idx1, idx2, off1, off2, loc1, loc2);
  me_laux       <<<1,     256, 0, stream>>>(gates, cnt1, out);
  scatter_out   <<<32,    256, 0, stream>>>(idx1, idx2, loc1, loc2, g1v, g2v, out);
}